// LuongAttention_47029891891730
// MI455X (gfx1250) — compile-verified
//
#include <hip/hip_runtime.h>
#include <hip/hip_bf16.h>

typedef __attribute__((ext_vector_type(2))) float v2f;
typedef __attribute__((ext_vector_type(8))) float v8f;

#define S_LEN 1024
#define B_SZ  32
#define D_SZ  1024

// ---------------------------------------------------------------------------
// Stage 1: u[b,d] = sum_e h[b,e] * W[e,d]   (h: [32,1024], W: [1024,1024])
// WMMA f32 16x16x4. 2 M-tiles x 64 N-tiles = 128 tiles, 1 wave per tile,
// 8 waves per block -> 16 blocks. 256 K-steps per tile.
//
// A 16x4 f32 layout (per ISA): VGPR0: lanes0-15 K=k+0, lanes16-31 K=k+2;
//                              VGPR1: lanes0-15 K=k+1, lanes16-31 K=k+3.
// => per lane a contiguous v2f at h[row][k + 2*half].
// B 4x16 f32 layout (rows striped across lanes within a VGPR):
//   VGPR0: K=k+2*half row, N=n0+l15 ; VGPR1: K=k+2*half+1 row.
// C/D 16x16 f32: VGPR v -> row m0 + v + 8*half, col n0 + l15.
// ---------------------------------------------------------------------------
__global__ __launch_bounds__(256) void gemm_u_kernel(const float* __restrict__ h,
                                                     const float* __restrict__ W,
                                                     float* __restrict__ u) {
    const int wave = threadIdx.x >> 5;
    const int lane = threadIdx.x & 31;
    const int tile = blockIdx.x * 8 + wave;   // 0..127
    const int m0 = (tile & 1) * 16;           // 0 or 16
    const int n0 = (tile >> 1) * 16;          // 0..1008
    const int half = lane >> 4;               // 0/1
    const int l15  = lane & 15;

    const float* hRow = h + (m0 + l15) * D_SZ;
    v8f acc = {};
    for (int k = 0; k < D_SZ; k += 4) {
        const int ka = k + (half << 1);
        v2f a = *(const v2f*)(hRow + ka);
        v2f b;
        b.x = W[(size_t)(ka + 0) * D_SZ + n0 + l15];
        b.y = W[(size_t)(ka + 1) * D_SZ + n0 + l15];
        acc = __builtin_amdgcn_wmma_f32_16x16x4_f32(
            /*neg_a=*/false, a, /*neg_b=*/false, b,
            /*c_mod=*/(short)0, acc, /*reuse_a=*/false, /*reuse_b=*/false);
    }
    const int mBase = m0 + half * 8;
    const int col = n0 + l15;
#pragma unroll
    for (int v = 0; v < 8; ++v)
        u[(size_t)(mBase + v) * D_SZ + col] = acc[v];
}

// ---------------------------------------------------------------------------
// Stage 1b: c[b] = dot(h[b,:], bias)   (bias is zero in the harness, kept for
// fidelity). One wave32 per b.
// ---------------------------------------------------------------------------
__global__ __launch_bounds__(32) void bias_dot_kernel(const float* __restrict__ h,
                                                      const float* __restrict__ bias,
                                                      float* __restrict__ c) {
    const int b = blockIdx.x;
    const int lane = threadIdx.x;
    float acc = 0.f;
    for (int d = lane; d < D_SZ; d += 32)
        acc += h[(size_t)b * D_SZ + d] * bias[d];
#pragma unroll
    for (int off = 16; off > 0; off >>= 1)
        acc += __shfl_xor(acc, off, 32);
    if (lane == 0) c[b] = acc;
}

// ---------------------------------------------------------------------------
// Stage 2: attn[b,s] = dot(enc[s,b,:], u[b,:]) + c[b]
// One wave per (s,b). 8 waves/block cover 8 consecutive b at one s ->
// contiguous 32KB of enc per block. float4 loads, wave32 shuffle reduce.
// ---------------------------------------------------------------------------
__global__ __launch_bounds__(256) void attn_kernel(const float* __restrict__ enc,
                                                   const float* __restrict__ u,
                                                   const float* __restrict__ c,
                                                   float* __restrict__ attn) {
    const int w = blockIdx.x * 8 + (threadIdx.x >> 5);   // 0..32767
    const int lane = threadIdx.x & 31;
    const int b = w & (B_SZ - 1);
    const int s = w >> 5;
    const float* e  = enc + ((size_t)s * B_SZ + b) * D_SZ;
    const float* ub = u + (size_t)b * D_SZ;
    float acc = 0.f;
#pragma unroll
    for (int i = 0; i < 8; ++i) {
        const int d = (i * 32 + lane) * 4;
        const float4 ev = *(const float4*)(e + d);
        const float4 uv = *(const float4*)(ub + d);
        acc += ev.x * uv.x + ev.y * uv.y + ev.z * uv.z + ev.w * uv.w;
    }
#pragma unroll
    for (int off = 16; off > 0; off >>= 1)
        acc += __shfl_xor(acc, off, 32);
    if (lane == 0) attn[(size_t)b * S_LEN + s] = acc + c[b];
}

// ---------------------------------------------------------------------------
// Stage 3: softmax over s for each b. One block (256 thr) per b, 4 vals/thr.
// Writes alpha into d_out[32768 + b*1024 + s].
// ---------------------------------------------------------------------------
__global__ __launch_bounds__(256) void softmax_kernel(const float* __restrict__ attn,
                                                      float* __restrict__ alpha) {
    __shared__ float sred[8];
    const int b = blockIdx.x;
    const int t = threadIdx.x;
    const float* row = attn + (size_t)b * S_LEN;
    float v0 = row[t], v1 = row[t + 256], v2 = row[t + 512], v3 = row[t + 768];

    float m = fmaxf(fmaxf(v0, v1), fmaxf(v2, v3));
#pragma unroll
    for (int off = 16; off > 0; off >>= 1)
        m = fmaxf(m, __shfl_xor(m, off, 32));
    if ((t & 31) == 0) sred[t >> 5] = m;
    __syncthreads();
    float mAll = sred[0];
#pragma unroll
    for (int i = 1; i < 8; ++i) mAll = fmaxf(mAll, sred[i]);
    __syncthreads();

    const float e0 = expf(v0 - mAll), e1 = expf(v1 - mAll);
    const float e2 = expf(v2 - mAll), e3 = expf(v3 - mAll);
    float sum = e0 + e1 + e2 + e3;
#pragma unroll
    for (int off = 16; off > 0; off >>= 1)
        sum += __shfl_xor(sum, off, 32);
    if ((t & 31) == 0) sred[t >> 5] = sum;
    __syncthreads();
    float sAll = 0.f;
#pragma unroll
    for (int i = 0; i < 8; ++i) sAll += sred[i];
    const float inv = 1.f / sAll;

    float* out = alpha + (size_t)b * S_LEN;
    out[t]       = e0 * inv;
    out[t + 256] = e1 * inv;
    out[t + 512] = e2 * inv;
    out[t + 768] = e3 * inv;
}

// ---------------------------------------------------------------------------
// Stage 4: partial[sc,b,d] = sum_{s in chunk sc} alpha[b,s] * enc[s,b,d]
// grid (schunk=8, b=32); 256 thr, each owns 4 d via float4. 512KB enc/block.
// ---------------------------------------------------------------------------
__global__ __launch_bounds__(256) void reps_partial_kernel(const float* __restrict__ enc,
                                                           const float* __restrict__ alpha,
                                                           float* __restrict__ partial) {
    const int sc = blockIdx.x;
    const int b  = blockIdx.y;
    const int d  = threadIdx.x * 4;
    float4 acc = {0.f, 0.f, 0.f, 0.f};
    const int s0 = sc * (S_LEN / 8);
    for (int s = s0; s < s0 + (S_LEN / 8); ++s) {
        const float a = alpha[(size_t)b * S_LEN + s];
        const float4 ev = *(const float4*)(enc + ((size_t)s * B_SZ + b) * D_SZ + d);
        acc.x += a * ev.x;
        acc.y += a * ev.y;
        acc.z += a * ev.z;
        acc.w += a * ev.w;
    }
    *(float4*)(partial + ((size_t)(sc * B_SZ + b)) * D_SZ + d) = acc;
}

// Stage 5: reps[b,d] = sum_sc partial[sc,b,d]  -> d_out[0 .. 32767]
__global__ __launch_bounds__(256) void reps_reduce_kernel(const float* __restrict__ partial,
                                                          float* __restrict__ reps) {
    const int idx = blockIdx.x * 256 + threadIdx.x;   // 0..32767
    float acc = 0.f;
#pragma unroll
    for (int sc = 0; sc < 8; ++sc)
        acc += partial[(size_t)sc * (B_SZ * D_SZ) + idx];
    reps[idx] = acc;
}

// ---------------------------------------------------------------------------
extern "C" void kernel_launch(void* const* d_in, const int* in_sizes, int n_in,
                              void* d_out, int out_size, void* d_ws, size_t ws_size,
                              hipStream_t stream) {
    const float* hidden = (const float*)d_in[0];   // [2,32,512] == flat [32,1024]
    const float* enc    = (const float*)d_in[1];   // [1024,32,1024]
    const float* W      = (const float*)d_in[2];   // [1024,1024]
    const float* bias   = (const float*)d_in[3];   // [1024]

    float* out = (float*)d_out;          // reps: [0,32768) ; alpha: [32768,65536)
    float* ws  = (float*)d_ws;
    float* u       = ws;                 // 32768 floats
    float* c       = ws + 32768;         // 32 floats
    float* attn    = ws + 32800;         // 32768 floats
    float* partial = ws + 65568;         // 8*32768 floats  (total ~1.25 MB)

    float* alpha = out + (B_SZ * D_SZ);  // second output half

    gemm_u_kernel<<<16, 256, 0, stream>>>(hidden, W, u);
    bias_dot_kernel<<<B_SZ, 32, 0, stream>>>(hidden, bias, c);
    attn_kernel<<<(S_LEN * B_SZ) / 8, 256, 0, stream>>>(enc, u, c, attn);
    softmax_kernel<<<B_SZ, 256, 0, stream>>>(attn, alpha);
    reps_partial_kernel<<<dim3(8, B_SZ), 256, 0, stream>>>(enc, alpha, partial);
    reps_reduce_kernel<<<(B_SZ * D_SZ) / 256, 256, 0, stream>>>(partial, out);
}